// GraphNetwork_86569360818954
// MI455X (gfx1250) — compile-verified
//
#include <hip/hip_runtime.h>
#include <hip/hip_bf16.h>

// ---------------------------------------------------------------------------
// GCN: h1 = relu(D^-1/2 (A+I) D^-1/2 (x W1) + b1); h2 likewise; out = h2 Wl + bl
// MI455X plan: fp32 WMMA (16x16x4) for the 3 GEMMs (FLOPs are negligible ->
// keep full precision); edge aggregation via L2-resident gathers (xw = 25.6MB
// << 192MB L2) and global_atomic_add_f32 scatter; degree/dinv computed once.
// ---------------------------------------------------------------------------

typedef float v2f __attribute__((ext_vector_type(2)));
typedef float v8f __attribute__((ext_vector_type(8)));

// ---------------- degree / dinv ----------------
__global__ void deg_init_kernel(float* __restrict__ deg, int n) {
    int i = blockIdx.x * blockDim.x + threadIdx.x;
    if (i < n) deg[i] = 1.0f;               // self loop contributes 1
}

__global__ void deg_count_kernel(const long long* __restrict__ dst,
                                 float* __restrict__ deg, long long e) {
    long long i = (long long)blockIdx.x * blockDim.x + threadIdx.x;
    if (i < e) atomicAdd(&deg[dst[i]], 1.0f);
}

__global__ void deg_to_dinv_kernel(float* __restrict__ deg, int n) {
    int i = blockIdx.x * blockDim.x + threadIdx.x;
    if (i < n) {
        float d = deg[i];
        deg[i] = (d > 0.0f) ? __frsqrt_rn(d) : 0.0f;
    }
}

// ---------------- fp32 WMMA GEMM: out[N,H] = A[N,K] @ W[K,H] (+ bias) -------
// One wave (32 lanes) computes one 16x16 tile of the output.
// ISA 7.12.2 fp32 layouts:
//   A 16x4 : lane l<16 -> row M=l, {v0,v1}={K,K+1}; lane>=16 -> {K+2,K+3}
//   B 4x16 : lane l -> col N=l%16, {v0,v1}={K,K+1} (halves K+2,K+3)
//   C 16x16: lane l -> col N=l%16, VGPR r -> row M=r (lanes 0-15) / r+8 (16-31)
template <int K, int H, bool FUSE_BIAS>
__global__ void gemm_wmma_f32_kernel(const float* __restrict__ A,
                                     const float* __restrict__ W,
                                     const float* __restrict__ bias,
                                     float* __restrict__ out,
                                     int nrows) {
    const int WAVES = 4;                              // 128 threads / block
    const int lane  = threadIdx.x & 31;
    const int wave  = threadIdx.x >> 5;
    const int tiles_n = H / 16;
    const int tiles_m = nrows / 16;
    const int tile = blockIdx.x * WAVES + wave;       // uniform per wave
    if (tile >= tiles_m * tiles_n) return;            // scalar branch only

    const int m0 = (tile / tiles_n) * 16;
    const int n0 = (tile % tiles_n) * 16;
    const int half = lane >> 4;                       // 0 or 1
    const int l    = lane & 15;

    v8f acc = {};
    const float* arow = A + (size_t)(m0 + l) * K;     // row for this lane
    #pragma unroll 4
    for (int k = 0; k < K; k += 4) {
        const int ka = k + half * 2;
        // A fragment: two consecutive K values -> 64-bit load
        v2f a;
        a.x = arow[ka];
        a.y = arow[ka + 1];
        // B fragment: W is row-major [K,H]
        v2f b;
        b.x = W[(size_t)ka * H + n0 + l];
        b.y = W[(size_t)(ka + 1) * H + n0 + l];
        acc = __builtin_amdgcn_wmma_f32_16x16x4_f32(
            /*neg_a=*/false, a, /*neg_b=*/false, b,
            /*c_mod=*/(short)0, acc, /*reuse_a=*/false, /*reuse_b=*/false);
    }

    const float bn = FUSE_BIAS ? bias[n0 + l] : 0.0f;
    #pragma unroll
    for (int r = 0; r < 8; ++r) {
        const int m = m0 + half * 8 + r;
        out[(size_t)m * H + n0 + l] = acc[r] + bn;
    }
}

// ---------------- aggregation (H fixed at 64) ------------------------------
// Self-loop term: out[i,:] = xw[i,:] * dinv[i]^2   (plain stores, runs first)
__global__ void self_loop_kernel(const float* __restrict__ xw,
                                 const float* __restrict__ dinv,
                                 float* __restrict__ out, int n) {
    int t = blockIdx.x * blockDim.x + threadIdx.x;    // n*16 threads, 4 cols ea
    int i = t >> 4;
    if (i >= n) return;
    int c = (t & 15) * 4;
    float di = dinv[i];
    float nrm = di * di;
    const float4 v = *reinterpret_cast<const float4*>(xw + (size_t)i * 64 + c);
    float4 o;
    o.x = v.x * nrm; o.y = v.y * nrm; o.z = v.z * nrm; o.w = v.w * nrm;
    *reinterpret_cast<float4*>(out + (size_t)i * 64 + c) = o;
}

// Edge term: out[dst,:] += xw[src,:] * dinv[src]*dinv[dst]
// 16 lanes per edge, float4 (b128) gather, 4x global_atomic_add_f32 scatter.
__global__ void agg_edges_kernel(const long long* __restrict__ src,
                                 const long long* __restrict__ dst,
                                 const float* __restrict__ dinv,
                                 const float* __restrict__ xw,
                                 float* __restrict__ out, long long e) {
    long long t = (long long)blockIdx.x * blockDim.x + threadIdx.x;
    long long ei = t >> 4;
    if (ei >= e) return;
    int c = (int)(t & 15) * 4;
    long long s = src[ei];
    long long d = dst[ei];
    float nrm = dinv[s] * dinv[d];
    const float4 v = *reinterpret_cast<const float4*>(xw + (size_t)s * 64 + c);
    float* o = out + (size_t)d * 64 + c;
    atomicAdd(o + 0, v.x * nrm);
    atomicAdd(o + 1, v.y * nrm);
    atomicAdd(o + 2, v.z * nrm);
    atomicAdd(o + 3, v.w * nrm);
}

// Bias + ReLU in place over [n,64]
__global__ void bias_relu_kernel(float* __restrict__ h,
                                 const float* __restrict__ b, int n) {
    int t = blockIdx.x * blockDim.x + threadIdx.x;
    if (t >= n * 64) return;
    float v = h[t] + b[t & 63];
    h[t] = fmaxf(v, 0.0f);
}

// ---------------------------------------------------------------------------
extern "C" void kernel_launch(void* const* d_in, const int* in_sizes, int n_in,
                              void* d_out, int out_size, void* d_ws, size_t ws_size,
                              hipStream_t stream) {
    const float*     x   = (const float*)d_in[0];
    const long long* ei  = (const long long*)d_in[1];
    const float*     W1  = (const float*)d_in[2];
    const float*     b1  = (const float*)d_in[3];
    const float*     W2  = (const float*)d_in[4];
    const float*     b2  = (const float*)d_in[5];
    const float*     Wl  = (const float*)d_in[6];
    const float*     bl  = (const float*)d_in[7];
    float*           out = (float*)d_out;

    const int       N = in_sizes[0] / 128;       // 100000
    const long long E = in_sizes[1] / 2;         // 3.2M
    const long long* src = ei;
    const long long* dst = ei + E;

    // Workspace: dinv[N] | bufA[N*64] | bufB[N*64]  (~52 MB)
    char*  ws   = (char*)d_ws;
    float* dinv = (float*)ws;
    size_t offA = ((size_t)N * sizeof(float) + 255) & ~(size_t)255;
    float* bufA = (float*)(ws + offA);
    float* bufB = bufA + (size_t)N * 64;

    const int TPB = 256;
    const int nblk_node  = (N + TPB - 1) / TPB;
    const int nblk_edge  = (int)((E + TPB - 1) / TPB);
    const int nblk_n16   = (int)(((long long)N * 16 + TPB - 1) / TPB);
    const int nblk_e16   = (int)((E * 16 + TPB - 1) / TPB);
    const int nblk_n64   = (int)(((long long)N * 64 + TPB - 1) / TPB);

    // 1) degree -> dinv (shared by both GCN layers)
    deg_init_kernel<<<nblk_node, TPB, 0, stream>>>(dinv, N);
    deg_count_kernel<<<nblk_edge, TPB, 0, stream>>>(dst, dinv, E);
    deg_to_dinv_kernel<<<nblk_node, TPB, 0, stream>>>(dinv, N);

    // GEMM grids: one wave per 16x16 tile, 4 waves (128 thr) per block
    const int tiles1 = (N / 16) * (64 / 16);
    const int tiles3 = (N / 16) * (32 / 16);
    const int gblk1  = (tiles1 + 3) / 4;
    const int gblk3  = (tiles3 + 3) / 4;

    // 2) layer 1: xw = x @ W1 ; aggregate ; +b1 ; relu
    gemm_wmma_f32_kernel<128, 64, false><<<gblk1, 128, 0, stream>>>(x, W1, nullptr, bufA, N);
    self_loop_kernel<<<nblk_n16, TPB, 0, stream>>>(bufA, dinv, bufB, N);
    agg_edges_kernel<<<nblk_e16, TPB, 0, stream>>>(src, dst, dinv, bufA, bufB, E);
    bias_relu_kernel<<<nblk_n64, TPB, 0, stream>>>(bufB, b1, N);

    // 3) layer 2: xw = h1 @ W2 ; aggregate ; +b2 ; relu
    gemm_wmma_f32_kernel<64, 64, false><<<gblk1, 128, 0, stream>>>(bufB, W2, nullptr, bufA, N);
    self_loop_kernel<<<nblk_n16, TPB, 0, stream>>>(bufA, dinv, bufB, N);
    agg_edges_kernel<<<nblk_e16, TPB, 0, stream>>>(src, dst, dinv, bufA, bufB, E);
    bias_relu_kernel<<<nblk_n64, TPB, 0, stream>>>(bufB, b2, N);

    // 4) head: out = h2 @ Wl + bl  (bias fused into WMMA epilogue)
    gemm_wmma_f32_kernel<64, 32, true><<<gblk3, 128, 0, stream>>>(bufB, Wl, bl, out, N);
}